// Decoder_39015482916866
// MI455X (gfx1250) — compile-verified
//
#include <hip/hip_runtime.h>
#include <hip/hip_bf16.h>
#include <math.h>

typedef float v2f __attribute__((ext_vector_type(2)));
typedef float v8f __attribute__((ext_vector_type(8)));

#define DD   1024
#define SS   2048
#define BB   64
#define EE   512
#define VV   32000

#if defined(__has_builtin)
#if __has_builtin(__builtin_amdgcn_global_load_async_to_lds_b32) && \
    __has_builtin(__builtin_amdgcn_s_wait_asynccnt)
#define USE_ASYNC_LDS 1
#endif
#endif
#ifndef USE_ASYNC_LDS
#define USE_ASYNC_LDS 0
#endif

#if USE_ASYNC_LDS
typedef __attribute__((address_space(1))) int g_i32;   // global (device) word
typedef __attribute__((address_space(3))) int l_i32;   // LDS word
#endif

// hardware tanh (CDNA5 V_TANH_F32 trans op); 2 v_nops cover the trans-op
// result-use hazard (ISA 7.4: 1 independent op or V_NOP after trans op).
__device__ __forceinline__ float tanh_hw(float x) {
  float y;
  asm volatile("v_tanh_f32 %0, %1\n\tv_nop\n\tv_nop" : "=v"(y) : "v"(x));
  return y;
}
__device__ __forceinline__ float sigmoid_fast(float x) {
  return 1.0f / (1.0f + __expf(-x));
}

// Stage a 16-row x 1024-col fp32 tile, TRANSPOSED, into As[1024][16].
// dst layout: As[c][r].  src rows are `ld` floats apart starting at `src`.
__device__ __forceinline__ void stage_tile_T(float (*As)[16],
                                             const float* __restrict__ src,
                                             int ld, int t) {
#if USE_ASYNC_LDS
  // async element-scatter: lanes read 32 consecutive floats of one row
  // (coalesced 128B) and scatter to transposed LDS slots. ASYNCcnt-tracked.
  #pragma unroll
  for (int i = 0; i < 64; ++i) {
    const int e = i * 256 + t;            // 16384 elements
    const int r = e >> 10, c = e & 1023;
    __builtin_amdgcn_global_load_async_to_lds_b32(
        (g_i32*)(src + (size_t)r * ld + c),
        (l_i32*)(&As[c][r]), 0, 0);
  }
  __builtin_amdgcn_s_wait_asynccnt(0);
#else
  const int r  = t & 15;
  const int c0 = (t >> 4) * 4;
  const float* s = src + (size_t)r * ld;
  #pragma unroll
  for (int i = 0; i < 16; ++i) {
    int c = c0 + i * 64;
    float4 v = *(const float4*)(s + c);
    As[c + 0][r] = v.x; As[c + 1][r] = v.y;
    As[c + 2][r] = v.z; As[c + 3][r] = v.w;
  }
#endif
}

// ---------------------------------------------------------------------------
// q = hidden @ W1_k + W1_b    (64 x 1024)
// ---------------------------------------------------------------------------
__global__ __launch_bounds__(256) void k_q(const float* __restrict__ hidden,
                                           const float* __restrict__ W1,
                                           const float* __restrict__ W1b,
                                           float* __restrict__ q) {
  int gid = blockIdx.x * 256 + threadIdx.x;   // 65536 total
  int b = gid >> 10, n = gid & 1023;
  float acc = W1b[n];
  const float* hb = hidden + b * DD;
  const float* w  = W1 + n;
  for (int k = 0; k < DD; ++k) acc += hb[k] * w[(size_t)k * DD];
  q[gid] = acc;
}

// ---------------------------------------------------------------------------
// Fused: score[b,s] = tanh(q[b,:] + enc[b,s,:]@W2 + W2b) . V_k  + V_b
// One WG = 16 rows of flattened (b,s). A staged transposed in LDS (64KB).
// 8 waves sweep disjoint N-blocks of 16; K-loop uses V_WMMA_F32_16X16X4_F32.
// ---------------------------------------------------------------------------
__global__ __launch_bounds__(256) void k_attn_scores(
    const float* __restrict__ enc, const float* __restrict__ W2,
    const float* __restrict__ W2b, const float* __restrict__ q,
    const float* __restrict__ Vk,  const float* __restrict__ Vb,
    float* __restrict__ score) {
  __shared__ float As[DD][16];                 // As[k][row] : 64 KB exactly
  const int m0   = blockIdx.x * 16;            // flattened (b,s) base
  const int b    = m0 >> 11;                   // /2048 (16 | 2048 so same b)
  const int t    = threadIdx.x;
  const int lane = t & 31, wave = t >> 5;

  stage_tile_T(As, enc + (size_t)m0 * DD, DD, t);
  __syncthreads();

  const int hl  = lane >> 4;                   // half of wave (0/1)
  const int l16 = lane & 15;
  float par[8];
  #pragma unroll
  for (int r = 0; r < 8; ++r) par[r] = 0.0f;

  for (int nb = wave * 16; nb < DD; nb += 128) {
    const int n = nb + l16;
    v8f acc = {0.f,0.f,0.f,0.f,0.f,0.f,0.f,0.f};
    const float* w2col = W2 + n;               // column n, stride D (L2 resident)
    #pragma unroll 4
    for (int k0 = 0; k0 < DD; k0 += 4) {
      const int kk = k0 + 2 * hl;
      v2f a;  a.x  = As[kk][l16];              a.y  = As[kk + 1][l16];
      v2f bv; bv.x = w2col[(size_t)kk * DD];   bv.y = w2col[(size_t)(kk + 1) * DD];
      acc = __builtin_amdgcn_wmma_f32_16x16x4_f32(false, a, false, bv,
                                                  (short)0, acc, false, false);
    }
    const float qn = q[b * DD + n] + W2b[n];
    const float vk = Vk[n];
    #pragma unroll
    for (int r = 0; r < 8; ++r) par[r] += tanh_hw(acc[r] + qn) * vk;
  }

  // reduce over the 16 lanes of each half-wave (wave32)
  #pragma unroll
  for (int r = 0; r < 8; ++r) {
    #pragma unroll
    for (int off = 8; off >= 1; off >>= 1)
      par[r] += __shfl_xor(par[r], off, 16);
  }
  __syncthreads();                             // A tile dead -> reuse as scratch
  float* red = &As[0][0];                      // red[wave*16 + row]
  if (l16 == 0) {
    #pragma unroll
    for (int r = 0; r < 8; ++r) red[wave * 16 + hl * 8 + r] = par[r];
  }
  __syncthreads();
  if (t < 16) {
    float s = 0.0f;
    #pragma unroll
    for (int w = 0; w < 8; ++w) s += red[w * 16 + t];
    score[m0 + t] = s + Vb[0];
  }
}

// ---------------------------------------------------------------------------
// softmax over S per batch row; writes attn output (B,S,1)
// ---------------------------------------------------------------------------
__global__ __launch_bounds__(256) void k_softmax(const float* __restrict__ score,
                                                 float* __restrict__ attn) {
  __shared__ float red[256];
  const int b = blockIdx.x, t = threadIdx.x;
  const float* s = score + b * SS;
  float v[8];
  float mx = -1e30f;
  #pragma unroll
  for (int i = 0; i < 8; ++i) { v[i] = s[t + i * 256]; mx = fmaxf(mx, v[i]); }
  red[t] = mx; __syncthreads();
  for (int o = 128; o >= 1; o >>= 1) {
    if (t < o) red[t] = fmaxf(red[t], red[t + o]);
    __syncthreads();
  }
  mx = red[0]; __syncthreads();
  float sum = 0.0f;
  #pragma unroll
  for (int i = 0; i < 8; ++i) { v[i] = __expf(v[i] - mx); sum += v[i]; }
  red[t] = sum; __syncthreads();
  for (int o = 128; o >= 1; o >>= 1) {
    if (t < o) red[t] += red[t + o];
    __syncthreads();
  }
  const float inv = 1.0f / red[0];
  #pragma unroll
  for (int i = 0; i < 8; ++i) attn[b * SS + t + i * 256] = v[i] * inv;
}

// ---------------------------------------------------------------------------
// partial context: 8 s-chunks per b, deterministic (no atomics)
// part[sc][b][d] = sum over 256 s of attn*enc
// ---------------------------------------------------------------------------
__global__ __launch_bounds__(256) void k_context_part(const float* __restrict__ enc,
                                                      const float* __restrict__ attn,
                                                      float* __restrict__ part) {
  const int b  = blockIdx.x >> 3;
  const int sc = blockIdx.x & 7;
  const int t  = threadIdx.x;                  // d = 4*t
  const float* a = attn + b * SS + sc * 256;
  const float* e = enc + ((size_t)b * SS + sc * 256) * DD + t * 4;
  float4 acc = {0.f, 0.f, 0.f, 0.f};
  for (int s = 0; s < 256; ++s) {
    __builtin_prefetch(e + (size_t)(s + 8) * DD, 0, 1);   // global_prefetch_b8
    const float av = a[s];
    float4 ev = *(const float4*)(e + (size_t)s * DD);
    acc.x += av * ev.x; acc.y += av * ev.y;
    acc.z += av * ev.z; acc.w += av * ev.w;
  }
  *(float4*)(part + ((size_t)sc * BB + b) * DD + t * 4) = acc;
}

__global__ __launch_bounds__(256) void k_ctx_reduce(const float* __restrict__ part,
                                                    float* __restrict__ ctx) {
  int i = blockIdx.x * 256 + threadIdx.x;      // 65536
  float s = 0.0f;
  #pragma unroll
  for (int c = 0; c < 8; ++c) s += part[(size_t)c * (BB * DD) + i];
  ctx[i] = s;
}

// ---------------------------------------------------------------------------
// LSTM cell: z = [ctx, emb[x]] @ lstm_k + hidden @ lstm_rk + b, gates fused
// ---------------------------------------------------------------------------
__global__ __launch_bounds__(256) void k_lstm(
    const float* __restrict__ ctx, const int* __restrict__ x,
    const float* __restrict__ emb, const float* __restrict__ hidden,
    const float* __restrict__ cell, const float* __restrict__ lk,
    const float* __restrict__ lrk, const float* __restrict__ lb,
    float* __restrict__ h_out, float* __restrict__ c_out) {
  const int gid = blockIdx.x * 256 + threadIdx.x;   // 65536
  const int b = gid >> 10, d = gid & 1023;
  float z0 = lb[d], z1 = lb[d + 1024], z2 = lb[d + 2048], z3 = lb[d + 3072];
  const float* cb = ctx + b * DD;
  for (int k = 0; k < DD; ++k) {
    const float xv = cb[k];
    const float* w = lk + (size_t)k * 4096 + d;
    z0 += xv * w[0]; z1 += xv * w[1024]; z2 += xv * w[2048]; z3 += xv * w[3072];
  }
  const float* er = emb + (size_t)x[b] * EE;
  for (int k = 0; k < EE; ++k) {
    const float xv = er[k];
    const float* w = lk + (size_t)(DD + k) * 4096 + d;
    z0 += xv * w[0]; z1 += xv * w[1024]; z2 += xv * w[2048]; z3 += xv * w[3072];
  }
  const float* hb = hidden + b * DD;
  for (int k = 0; k < DD; ++k) {
    const float hv = hb[k];
    const float* w = lrk + (size_t)k * 4096 + d;
    z0 += hv * w[0]; z1 += hv * w[1024]; z2 += hv * w[2048]; z3 += hv * w[3072];
  }
  const float ig = sigmoid_fast(z0), fg = sigmoid_fast(z1);
  const float gg = tanh_hw(z2),      og = sigmoid_fast(z3);
  const float cn = fg * cell[gid] + ig * gg;
  c_out[gid] = cn;
  h_out[gid] = og * tanh_hw(cn);
}

// ---------------------------------------------------------------------------
// prediction = h_new @ fc_k + fc_b  via WMMA.  Grid: n-tile(250) x m-tile(4),
// m inner so consecutive blocks share the fc_k n-panel in L2.
// ---------------------------------------------------------------------------
__global__ __launch_bounds__(256) void k_fc(const float* __restrict__ h,
                                            const float* __restrict__ fc,
                                            const float* __restrict__ fcb,
                                            float* __restrict__ pred) {
  __shared__ float As[DD][16];                 // 64 KB transposed h tile
  const int mt = blockIdx.x & 3;
  const int nt = blockIdx.x >> 2;
  const int t = threadIdx.x, lane = t & 31, wave = t >> 5;
  const int m0 = mt * 16;

  stage_tile_T(As, h + (size_t)m0 * DD, DD, t);
  __syncthreads();

  const int hl = lane >> 4, l16 = lane & 15;
  const int n = nt * 128 + wave * 16 + l16;
  v8f acc = {0.f,0.f,0.f,0.f,0.f,0.f,0.f,0.f};
  const float* col = fc + n;
  #pragma unroll 4
  for (int k0 = 0; k0 < DD; k0 += 4) {
    const int kk = k0 + 2 * hl;
    v2f a;  a.x  = As[kk][l16];               a.y  = As[kk + 1][l16];
    v2f bv; bv.x = col[(size_t)kk * VV];      bv.y = col[(size_t)(kk + 1) * VV];
    acc = __builtin_amdgcn_wmma_f32_16x16x4_f32(false, a, false, bv,
                                                (short)0, acc, false, false);
  }
  const float bias = fcb[n];
  #pragma unroll
  for (int r = 0; r < 8; ++r) {
    const int m = m0 + hl * 8 + r;
    pred[(size_t)m * VV + n] = acc[r] + bias;
  }
}

// ---------------------------------------------------------------------------
extern "C" void kernel_launch(void* const* d_in, const int* in_sizes, int n_in,
                              void* d_out, int out_size, void* d_ws, size_t ws_size,
                              hipStream_t stream) {
  const int*   x      = (const int*)  d_in[0];
  const float* hidden = (const float*)d_in[1];
  const float* cell   = (const float*)d_in[2];
  const float* enc    = (const float*)d_in[3];
  const float* W1k    = (const float*)d_in[4];
  const float* W1b    = (const float*)d_in[5];
  const float* W2k    = (const float*)d_in[6];
  const float* W2b    = (const float*)d_in[7];
  const float* Vk     = (const float*)d_in[8];
  const float* Vb     = (const float*)d_in[9];
  const float* emb    = (const float*)d_in[10];
  const float* lk     = (const float*)d_in[11];
  const float* lrk    = (const float*)d_in[12];
  const float* lb     = (const float*)d_in[13];
  const float* fck    = (const float*)d_in[14];
  const float* fcb    = (const float*)d_in[15];

  float* out  = (float*)d_out;
  float* pred = out;                                     // 64*32000
  float* hout = out + (size_t)BB * VV;                   // 64*1024
  float* cout = hout + (size_t)BB * DD;                  // 64*1024
  float* attn = cout + (size_t)BB * DD;                  // 64*2048

  float* ws    = (float*)d_ws;
  float* q     = ws;                                     // 65536
  float* score = q + (size_t)BB * DD;                    // 131072
  float* part  = score + (size_t)BB * SS;                // 8*65536
  float* ctx   = part + (size_t)8 * BB * DD;             // 65536

  k_q           <<<(BB * DD) / 256, 256, 0, stream>>>(hidden, W1k, W1b, q);
  k_attn_scores <<<(BB * SS) / 16, 256, 0, stream>>>(enc, W2k, W2b, q, Vk, Vb, score);
  k_softmax     <<<BB, 256, 0, stream>>>(score, attn);
  k_context_part<<<BB * 8, 256, 0, stream>>>(enc, attn, part);
  k_ctx_reduce  <<<(BB * DD) / 256, 256, 0, stream>>>(part, ctx);
  k_lstm        <<<(BB * DD) / 256, 256, 0, stream>>>(ctx, x, emb, hidden, cell,
                                                      lk, lrk, lb, hout, cout);
  k_fc          <<<(VV / 128) * 4, 256, 0, stream>>>(hout, fck, fcb, pred);
}